// NodeFeatureLayer_2001454760695
// MI455X (gfx1250) — compile-verified
//
#include <hip/hip_runtime.h>
#include <math.h>

#define EPS 1e-20f

typedef __attribute__((ext_vector_type(2))) float v2f;
typedef __attribute__((ext_vector_type(8))) float v8f;

// ---------------------------------------------------------------------------
// Kernel 1: Ux = x @ U_w^T + U_b  -> out
//           Vx = x @ V_w^T + V_b  -> ws
// One wave per 16x16 output tile, U and V accumulated together (A reused).
// Exact-fp32 V_WMMA_F32_16X16X4_F32 (no precision loss vs reference).
// HT > 0: compile-time H (shift addressing, fully unrollable K-loop).
// HT == 0: runtime H fallback.
// ---------------------------------------------------------------------------
template <int HT>
__global__ __launch_bounds__(32)
void gemm_uv_wmma_f32(const float* __restrict__ x,
                      const float* __restrict__ Uw, const float* __restrict__ Ub,
                      const float* __restrict__ Vw, const float* __restrict__ Vb,
                      float* __restrict__ Ux, float* __restrict__ Vx, int h_rt)
{
    const int H    = HT > 0 ? HT : h_rt;
    const int lane = threadIdx.x;        // 0..31, EXEC all ones (WMMA requirement)
    const int m    = lane & 15;          // row (A) / col (B) within tile
    const int kh   = (lane >> 4) << 1;   // K sub-offset: 0 or 2
    const int row0 = blockIdx.x * 16;
    const int col0 = blockIdx.y * 16;

    const float* xr = x  + (size_t)(row0 + m) * H;
    const float* ur = Uw + (size_t)(col0 + m) * H;
    const float* vr = Vw + (size_t)(col0 + m) * H;

    v8f accU = {};
    v8f accV = {};

    #pragma unroll 8
    for (int k = 0; k < H; k += 4) {
        v2f a  = *(const v2f*)(xr + k + kh);
        v2f bu = *(const v2f*)(ur + k + kh);
        v2f bv = *(const v2f*)(vr + k + kh);
        // 8 args: (neg_a, A, neg_b, B, c_mod, C, reuse_a, reuse_b)
        accU = __builtin_amdgcn_wmma_f32_16x16x4_f32(false, a, false, bu,
                                                     (short)0, accU, false, false);
        accV = __builtin_amdgcn_wmma_f32_16x16x4_f32(false, a, false, bv,
                                                     (short)0, accV, false, false);
    }

    // C/D f32 16x16 layout: VGPR g -> row M = g (+8 for lanes 16-31), col N = lane&15
    const int col   = col0 + m;
    const float bU  = Ub[col];
    const float bV  = Vb[col];
    const int rbase = row0 + ((lane >> 4) << 3);
    #pragma unroll
    for (int g = 0; g < 8; ++g) {
        const size_t idx = (size_t)(rbase + g) * H + col;
        Ux[idx] = accU[g] + bU;
        Vx[idx] = accV[g] + bV;
    }
}

// ---------------------------------------------------------------------------
// Kernel 2: for each (b, i):
//   num[h] = sum_j gate[b,i,j,h] * Vx[b,j,h]
//   den[h] = sum_j gate[b,i,j,h]
//   out[b,i,h] += num[h] / (EPS + den[h])      (out already holds Ux)
//
// One block per (b,i); the (N,H) gate slab is one contiguous stream, consumed
// with float4 (global_load_b128) loads. 1024 % H == 0 so each thread's
// 4-channel group is fixed for the whole loop -> register-only accumulation.
// Vx (2 MB total) stays resident in the 192 MB L2.
// ---------------------------------------------------------------------------
template <int NT, int HT>
__global__ __launch_bounds__(256)
void gate_reduce(const float* __restrict__ gate,  // (B, N, N, H)
                 const float* __restrict__ Vx,    // (B, N, H)
                 float* __restrict__ out,         // (B, N, H), holds Ux on entry
                 int n_rt, int h_rt)
{
    const int N  = NT > 0 ? NT : n_rt;
    const int H  = HT > 0 ? HT : h_rt;
    const int bi = blockIdx.x;           // b*N + i
    const int b  = bi / N;

    const float* g  = gate + (size_t)bi * N * H;   // 64K floats for (N=512,H=128)
    const float* vx = Vx   + (size_t)b  * N * H;

    const int t   = threadIdx.x;
    const int c0  = (4 * t) % H;         // fixed 4-channel group for this thread
    const int j0  = (4 * t) / H;
    const int jst = (4 * 256) / H;       // j advance per step (=8 for H=128)

    float n0 = 0.f, n1 = 0.f, n2 = 0.f, n3 = 0.f;
    float d0 = 0.f, d1 = 0.f, d2 = 0.f, d3 = 0.f;

    const int total = N * H;
    #pragma unroll 4
    for (int f = 4 * t, j = j0; f < total; f += 4 * 256, j += jst) {
        const float4 gg = *(const float4*)(g + f);
        const float4 vv = *(const float4*)(vx + (size_t)j * H + c0);
        // speculative prefetch ~32KB ahead (silently dropped if OOB)
        __builtin_prefetch(g + f + 8192, 0, 0);
        n0 = fmaf(gg.x, vv.x, n0);
        n1 = fmaf(gg.y, vv.y, n1);
        n2 = fmaf(gg.z, vv.z, n2);
        n3 = fmaf(gg.w, vv.w, n3);
        d0 += gg.x;
        d1 += gg.y;
        d2 += gg.z;
        d3 += gg.w;
    }

    // 8 threads (t, t+32, ..., t+224) share the same channel group -> LDS reduce
    __shared__ float4 sN[256];
    __shared__ float4 sD[256];
    sN[t] = make_float4(n0, n1, n2, n3);
    sD[t] = make_float4(d0, d1, d2, d3);
    __syncthreads();

    if (t < 32) {
        float4 N4 = sN[t];
        float4 D4 = sD[t];
        #pragma unroll
        for (int r = 1; r < 8; ++r) {
            const float4 a = sN[t + 32 * r];
            const float4 d = sD[t + 32 * r];
            N4.x += a.x; N4.y += a.y; N4.z += a.z; N4.w += a.w;
            D4.x += d.x; D4.y += d.y; D4.z += d.z; D4.w += d.w;
        }
        float* o = out + (size_t)bi * H + 4 * t;
        o[0] += N4.x / (EPS + D4.x);
        o[1] += N4.y / (EPS + D4.y);
        o[2] += N4.z / (EPS + D4.z);
        o[3] += N4.w / (EPS + D4.w);
    }
}

extern "C" void kernel_launch(void* const* d_in, const int* in_sizes, int n_in,
                              void* d_out, int out_size, void* d_ws, size_t ws_size,
                              hipStream_t stream)
{
    const float* x    = (const float*)d_in[0];   // (B, N, H)
    const float* gate = (const float*)d_in[1];   // (B, N, N, H)
    const float* Uw   = (const float*)d_in[2];   // (H, H)
    const float* Ub   = (const float*)d_in[3];   // (H,)
    const float* Vw   = (const float*)d_in[4];   // (H, H)
    const float* Vb   = (const float*)d_in[5];   // (H,)
    float* out        = (float*)d_out;           // (B, N, H)
    float* Vx         = (float*)d_ws;            // (B, N, H) scratch

    const int H  = in_sizes[3];                  // 128
    const int R  = in_sizes[0] / H;              // B*N = 4096
    const int N  = in_sizes[1] / in_sizes[0];    // 512
    (void)n_in; (void)out_size; (void)ws_size;

    dim3 g1(R / 16, H / 16);
    if (H == 128 && N == 512) {
        // Specialized: compile-time shapes -> shift addressing, clean unrolls.
        gemm_uv_wmma_f32<128><<<g1, 32, 0, stream>>>(x, Uw, Ub, Vw, Vb, out, Vx, H);
        gate_reduce<512, 128><<<R, 256, 0, stream>>>(gate, Vx, out, N, H);
    } else {
        // Generic fallback (runtime shapes).
        gemm_uv_wmma_f32<0><<<g1, 32, 0, stream>>>(x, Uw, Ub, Vw, Vb, out, Vx, H);
        gate_reduce<0, 0><<<R, 256, 0, stream>>>(gate, Vx, out, N, H);
    }
}